// Pooling_82205674045894
// MI455X (gfx1250) — compile-verified
//
#include <hip/hip_runtime.h>
#include <hip/hip_bf16.h>
#include <stdint.h>

// ---------------- problem constants ----------------
#define N_PTS   1000000
#define N_EDGES 4000000
#define CFEAT   64
#define M_VOX   8192          // 16 * 16 * 1 * 32

// ---------------- workspace layout (uint32 words) ----------------
#define W_RAW   0u            // [N] raw voxel id per point (reused as ORDER later)
#define W_ORDER 0u            // [N] point ids sorted by cluster (counting sort)
#define W_INV   1048576u      // [N] compact cluster id per point
#define W_OCC   2097152u      // [8192] occupancy flags
#define W_RANK  2105344u      // [8192] exclusive rank of each raw id
#define W_META  2113536u      // [16]   meta[0] = n_clusters
#define W_ICNT  2113552u      // [8192] int point count per cluster
#define W_POS   2121744u      // [8192*3] float position sums
#define W_BMAP  2146320u      // [2097152] 64M-bit edge-key bitmap
#define W_BLK   4243472u      // [256] per-block popcount totals
#define W_BOFF  4243728u      // [256] exclusive block offsets
#define W_OFF   4243984u      // [8192] cluster start offsets (read-only copy)
#define W_CUR   4252176u      // [8192] cluster scatter cursors (mutable)

// ---------------- output layout (float elements) ----------------
#define O_X     0u            // 8192*64
#define O_POS   524288u       // 8192*3
#define O_BATCH 548864u       // 8192
#define O_EDGE  557056u       // 2*4,000,000 (row0 then row1)
#define O_VALID 8557056u      // 8192
#define OUT_TOTAL 8565248u

// ---------------- gfx1250 feature probes ----------------
#if defined(__has_builtin)
#  if __has_builtin(__builtin_amdgcn_global_load_async_to_lds_b64)
#    define HAVE_ASYNC_B64 1
#  endif
#  if __has_builtin(__builtin_amdgcn_s_wait_asynccnt)
#    define HAVE_WAIT_ASYNC 1
#  endif
#  if __has_builtin(__builtin_amdgcn_s_wait_tensorcnt)
#    define HAVE_WAIT_TENSOR 1
#  endif
#endif

#if HAVE_ASYNC_B64
// parameter types follow the b128 pattern hipcc confirmed: vNi in addrspace(1)/(3)
typedef int v2i_async __attribute__((ext_vector_type(2)));
typedef __attribute__((address_space(1))) v2i_async* gptr_v2i;
typedef __attribute__((address_space(3))) v2i_async* lptr_v2i;
#endif

// ---------------- 0) initialize d_out ----------------
__global__ __launch_bounds__(256) void kInitOut(float* __restrict__ out) {
  unsigned i = blockIdx.x * 256u + threadIdx.x;
  if (i >= OUT_TOTAL) return;
  float v;
  if (i < O_EDGE)       v = 0.0f;    // x / pos / batch
  else if (i < O_VALID) v = -1.0f;   // edge padding
  else                  v = 0.0f;    // valid mask
  out[i] = v;
}

// ---------------- 1) voxel id + occupancy ----------------
__global__ __launch_bounds__(256) void kVoxelId(const float* __restrict__ pos,
                                                const int* __restrict__ batch,
                                                int* __restrict__ rawArr,
                                                int* __restrict__ occ) {
  int i = blockIdx.x * 256 + threadIdx.x;
  if (i >= N_PTS) return;
  __builtin_prefetch(&pos[(i + 2048) * 3], 0, 1);   // global_prefetch_b8
  float px = pos[i * 3 + 0];
  float py = pos[i * 3 + 1];
  int cx = min(max((int)floorf(px * 16.0f), 0), 15);  // voxel 1/16
  int cy = min(max((int)floorf(py * 16.0f), 0), 15);
  // t-dim grid is 1 -> clipped to 0; strides {1,16,256,256}
  int raw = cx + cy * 16 + batch[i] * 256;            // in [0, 8192)
  rawArr[i] = raw;
  occ[raw] = 1;                                        // benign race, all write 1
}

// ---------------- 2) scan 8192 occupancy flags -> rank table ----------------
__global__ __launch_bounds__(256) void kClusterScan(const int* __restrict__ occ,
                                                    int* __restrict__ rank,
                                                    int* __restrict__ meta) {
  __shared__ int sh[256];
  int tid = threadIdx.x;
  int base = tid * 32;
  int loc[32];
  int tot = 0;
#pragma unroll
  for (int j = 0; j < 32; ++j) { loc[j] = occ[base + j]; tot += loc[j]; }
  sh[tid] = tot;
  __syncthreads();
  for (int off = 1; off < 256; off <<= 1) {
    int v = (tid >= off) ? sh[tid - off] : 0;
    __syncthreads();
    sh[tid] += v;
    __syncthreads();
  }
  int run = sh[tid] - tot;
#pragma unroll
  for (int j = 0; j < 32; ++j) { rank[base + j] = run; run += loc[j]; }
  if (tid == 255) meta[0] = sh[255];  // number of occupied clusters
}

// ---------------- 3) per-point map: inv, counts, pos sums, batch ----------------
__global__ __launch_bounds__(256) void kPointMap(const float* __restrict__ pos,
                                                 const int* __restrict__ batch,
                                                 const int* __restrict__ rawArr,
                                                 const int* __restrict__ rank,
                                                 int* __restrict__ invArr,
                                                 int* __restrict__ icnt,
                                                 float* __restrict__ psum,
                                                 float* __restrict__ out) {
  int i = blockIdx.x * 256 + threadIdx.x;
  if (i >= N_PTS) return;
  int r = rank[rawArr[i]];
  invArr[i] = r;
  atomicAdd(&icnt[r], 1);
  atomicAdd(&psum[r * 3 + 0], pos[i * 3 + 0]);
  atomicAdd(&psum[r * 3 + 1], pos[i * 3 + 1]);
  atomicAdd(&psum[r * 3 + 2], pos[i * 3 + 2]);
  out[O_BATCH + (unsigned)r] = (float)batch[i];   // all writers agree per cluster
}

// ---------------- 4a) scan per-cluster counts -> start offsets + cursors -----
__global__ __launch_bounds__(256) void kOffsets(const int* __restrict__ icnt,
                                                int* __restrict__ off,
                                                int* __restrict__ cur) {
  __shared__ int sh[256];
  int tid = threadIdx.x;
  int base = tid * 32;
  int loc[32];
  int tot = 0;
#pragma unroll
  for (int j = 0; j < 32; ++j) { loc[j] = icnt[base + j]; tot += loc[j]; }
  sh[tid] = tot;
  __syncthreads();
  for (int o = 1; o < 256; o <<= 1) {
    int v = (tid >= o) ? sh[tid - o] : 0;
    __syncthreads();
    sh[tid] += v;
    __syncthreads();
  }
  int run = sh[tid] - tot;
#pragma unroll
  for (int j = 0; j < 32; ++j) {
    off[base + j] = run;
    cur[base + j] = run;
    run += loc[j];
  }
}

// ---------------- 4b) counting-sort scatter: point ids grouped by cluster ----
__global__ __launch_bounds__(256) void kScatter(const int* __restrict__ invArr,
                                                int* __restrict__ cur,
                                                int* __restrict__ order) {
  int i = blockIdx.x * 256 + threadIdx.x;
  if (i >= N_PTS) return;
  int slot = atomicAdd(&cur[invArr[i]], 1);
  order[slot] = i;
}

// ---------------- 4c) feature segment-max: one wave per cluster --------------
// Double-buffered async-to-LDS gather: while reducing point i from LDS, the
// async engine is already pulling point i+1's 256B row (8B per lane).
__global__ __launch_bounds__(256) void kFeatMaxGather(const float* __restrict__ x,
                                                      const int* __restrict__ order,
                                                      const int* __restrict__ off,
                                                      const int* __restrict__ icnt,
                                                      float* __restrict__ xout) {
  __shared__ float buf[8][2][CFEAT];   // per-wave private double buffer (4 KB)
  const int tid = threadIdx.x;
  const int w = tid >> 5;
  const int lane = tid & 31;
  const int c = blockIdx.x * 8 + w;    // cluster id, < 8192
  const int count = icnt[c];
  if (count == 0) return;              // wave-uniform; empty clusters untouched
  const int start = off[c];

  float ax = __int_as_float(0xFF800000);  // -inf
  float ay = ax;

#if HAVE_ASYNC_B64
  {
    int pid0 = order[start];
    __builtin_amdgcn_global_load_async_to_lds_b64(
        (gptr_v2i)(x + (size_t)pid0 * CFEAT + lane * 2),
        (lptr_v2i)&buf[w][0][lane * 2], 0, 0);
  }
  for (int i = 0; i < count; ++i) {
    if (i + 1 < count) {
      int pid = order[start + i + 1];
      __builtin_amdgcn_global_load_async_to_lds_b64(
          (gptr_v2i)(x + (size_t)pid * CFEAT + lane * 2),
          (lptr_v2i)&buf[w][(i + 1) & 1][lane * 2], 0, 0);
#  if HAVE_WAIT_ASYNC
      __builtin_amdgcn_s_wait_asynccnt(1);   // buffer i is now resident
#  else
      asm volatile("s_wait_asynccnt 1" ::: "memory");
#  endif
    } else {
#  if HAVE_WAIT_ASYNC
      __builtin_amdgcn_s_wait_asynccnt(0);
#  else
      asm volatile("s_wait_asynccnt 0" ::: "memory");
#  endif
    }
    const float2 v = *(const float2*)&buf[w][i & 1][lane * 2];
    ax = fmaxf(ax, v.x);
    ay = fmaxf(ay, v.y);
  }
#  if HAVE_WAIT_TENSOR
  __builtin_amdgcn_s_wait_tensorcnt(0);      // no TDM outstanding; documents path
#  endif
#else
  for (int i = 0; i < count; ++i) {
    int pid = order[start + i];
    const float2 v = *(const float2*)&x[(size_t)pid * CFEAT + lane * 2];
    ax = fmaxf(ax, v.x);
    ay = fmaxf(ay, v.y);
  }
#endif
  float2 res = make_float2(ax, ay);
  *(float2*)&xout[(size_t)c * CFEAT + lane * 2] = res;   // global_store_b64
}

// ---------------- 5) edge keys -> 64M-bit bitmap ----------------
__global__ __launch_bounds__(256) void kEdgeBitmap(const int* __restrict__ e0,
                                                   const int* __restrict__ e1,
                                                   const int* __restrict__ invArr,
                                                   unsigned int* __restrict__ bmap) {
  int e = blockIdx.x * 256 + threadIdx.x;
  if (e >= N_EDGES) return;
  __builtin_prefetch(&e0[e + 4096], 0, 1);
  int s = invArr[e0[e]];
  int d = invArr[e1[e]];
  if (s != d) {
    unsigned key = (unsigned)s * (unsigned)M_VOX + (unsigned)d;  // < 64M
    atomicOr(&bmap[key >> 5], 1u << (key & 31u));
  }
}

// ---------------- 6a) per-block popcount totals (256 blocks x 8192 words) ----
__global__ __launch_bounds__(256) void kEdgeBlockSum(const unsigned int* __restrict__ bmap,
                                                     int* __restrict__ blk) {
  __shared__ int sh[256];
  int tid = threadIdx.x;
  unsigned wbase = blockIdx.x * 8192u + (unsigned)tid * 32u;
  int c = 0;
#pragma unroll 4
  for (int j = 0; j < 32; ++j) c += __popc(bmap[wbase + j]);
  sh[tid] = c;
  __syncthreads();
  for (int off = 128; off > 0; off >>= 1) {
    if (tid < off) sh[tid] += sh[tid + off];
    __syncthreads();
  }
  if (tid == 0) blk[blockIdx.x] = sh[0];
}

// ---------------- 6b) exclusive scan of 256 block totals ----------------
__global__ __launch_bounds__(256) void kScan256(const int* __restrict__ blk,
                                                int* __restrict__ boff) {
  __shared__ int sh[256];
  int tid = threadIdx.x;
  int v = blk[tid];
  sh[tid] = v;
  __syncthreads();
  for (int off = 1; off < 256; off <<= 1) {
    int u = (tid >= off) ? sh[tid - off] : 0;
    __syncthreads();
    sh[tid] += u;
    __syncthreads();
  }
  boff[tid] = sh[tid] - v;
}

// ---------------- 6c) emit sorted, deduped edges by bit rank ----------------
__global__ __launch_bounds__(256) void kEdgeEmit(const unsigned int* __restrict__ bmap,
                                                 const int* __restrict__ boff,
                                                 float* __restrict__ out) {
  __shared__ int sh[256];
  int tid = threadIdx.x;
  unsigned wbase = blockIdx.x * 8192u + (unsigned)tid * 32u;
  int c = 0;
#pragma unroll 4
  for (int j = 0; j < 32; ++j) c += __popc(bmap[wbase + j]);
  sh[tid] = c;
  __syncthreads();
  for (int off = 1; off < 256; off <<= 1) {
    int u = (tid >= off) ? sh[tid - off] : 0;
    __syncthreads();
    sh[tid] += u;
    __syncthreads();
  }
  int rank = boff[blockIdx.x] + sh[tid] - c;   // exclusive global rank
  for (int j = 0; j < 32; ++j) {
    unsigned w = bmap[wbase + j];
    unsigned keybase = (wbase + j) * 32u;
    while (w) {
      int b = __ffs(w) - 1;
      unsigned key = keybase + (unsigned)b;
      out[O_EDGE + (unsigned)rank]             = (float)(key >> 13);     // key / 8192
      out[O_EDGE + 4000000u + (unsigned)rank]  = (float)(key & 8191u);   // key % 8192
      ++rank;
      w &= w - 1u;
    }
  }
}

// ---------------- 7) finalize clusters: mean pos, pixel round, valid ----------
__global__ __launch_bounds__(256) void kFinalize(const int* __restrict__ icnt,
                                                 const float* __restrict__ psum,
                                                 const int* __restrict__ meta,
                                                 float* __restrict__ out) {
  int r = blockIdx.x * 256 + threadIdx.x;
  if (r >= M_VOX) return;
  int n = meta[0];
  bool valid = r < n;
  float c = fmaxf((float)icnt[r], 1.0f);
  float px = psum[r * 3 + 0] / c;
  float py = psum[r * 3 + 1] / c;
  float pt = psum[r * 3 + 2] / c;
  float xq = floorf((px + 1e-5f) * 640.0f) * (1.0f / 640.0f);
  float yq = floorf((py + 1e-5f) * 480.0f) * (1.0f / 480.0f);
  out[O_POS + (unsigned)r * 3u + 0u] = valid ? xq : 0.0f;
  out[O_POS + (unsigned)r * 3u + 1u] = valid ? yq : 0.0f;
  out[O_POS + (unsigned)r * 3u + 2u] = valid ? pt : 0.0f;
  out[O_VALID + (unsigned)r] = valid ? 1.0f : 0.0f;
  if (!valid) {
    out[O_BATCH + (unsigned)r] = 0.0f;
#pragma unroll 4
    for (int f = 0; f < CFEAT; ++f) out[(unsigned)r * CFEAT + (unsigned)f] = 0.0f;
  }
}

extern "C" void kernel_launch(void* const* d_in, const int* in_sizes, int n_in,
                              void* d_out, int out_size, void* d_ws, size_t ws_size,
                              hipStream_t stream) {
  const float* x     = (const float*)d_in[0];
  const float* pos   = (const float*)d_in[1];
  const int*   batch = (const int*)d_in[2];
  const int*   eidx  = (const int*)d_in[3];
  const int*   e0    = eidx;
  const int*   e1    = eidx + N_EDGES;

  float*        out = (float*)d_out;
  unsigned int* ws  = (unsigned int*)d_ws;

  int*          rawArr = (int*)(ws + W_RAW);
  int*          order  = (int*)(ws + W_ORDER);   // reuses raw region (raw is dead)
  int*          invArr = (int*)(ws + W_INV);
  int*          occ    = (int*)(ws + W_OCC);
  int*          rank   = (int*)(ws + W_RANK);
  int*          meta   = (int*)(ws + W_META);
  int*          icnt   = (int*)(ws + W_ICNT);
  float*        psum   = (float*)(ws + W_POS);
  unsigned int* bmap   = ws + W_BMAP;
  int*          blk    = (int*)(ws + W_BLK);
  int*          boff   = (int*)(ws + W_BOFF);
  int*          coff   = (int*)(ws + W_OFF);
  int*          cur    = (int*)(ws + W_CUR);

  // zero occupancy / counts / pos sums / bitmap each call (graph-capture safe)
  (void)hipMemsetAsync(ws + W_OCC, 0, (size_t)(W_BLK - W_OCC) * 4u, stream);

  kInitOut<<<(OUT_TOTAL + 255u) / 256u, 256, 0, stream>>>(out);
  kVoxelId<<<(N_PTS + 255) / 256, 256, 0, stream>>>(pos, batch, rawArr, occ);
  kClusterScan<<<1, 256, 0, stream>>>(occ, rank, meta);
  kPointMap<<<(N_PTS + 255) / 256, 256, 0, stream>>>(pos, batch, rawArr, rank,
                                                     invArr, icnt, psum, out);
  kOffsets<<<1, 256, 0, stream>>>(icnt, coff, cur);
  kScatter<<<(N_PTS + 255) / 256, 256, 0, stream>>>(invArr, cur, order);
  kFeatMaxGather<<<M_VOX / 8, 256, 0, stream>>>(x, order, coff, icnt, out + O_X);
  kEdgeBitmap<<<(N_EDGES + 255) / 256, 256, 0, stream>>>(e0, e1, invArr, bmap);
  kEdgeBlockSum<<<256, 256, 0, stream>>>(bmap, blk);
  kScan256<<<1, 256, 0, stream>>>(blk, boff);
  kEdgeEmit<<<256, 256, 0, stream>>>(bmap, boff, out);
  kFinalize<<<M_VOX / 256, 256, 0, stream>>>(icnt, psum, meta, out);

  (void)in_sizes; (void)n_in; (void)out_size; (void)ws_size;
}